// Embed2GraphByLinear_28613072126235
// MI455X (gfx1250) — compile-verified
//
#include <hip/hip_runtime.h>
#include <stdint.h>

typedef __attribute__((ext_vector_type(2))) float v2f;
typedef __attribute__((ext_vector_type(8))) float v8f;

#define BB 8
#define NN 360
#define DD 128
#define NTILE 23   // ceil(360/16)
#define DTILE 8    // 128/16
#define TI 32
#define TJ 32
#define LSTR 132   // LDS row stride in floats (128 + 4 pad -> bank-conflict-free)

// ----- async global->LDS staging (builtin confirmed present by round-2 diag) -
typedef int v4i __attribute__((vector_size(4 * sizeof(int))));
typedef __attribute__((address_space(1))) v4i as1v4i;   // printed as "__device__" by clang/HIP
typedef __attribute__((address_space(3))) v4i as3v4i;   // LDS

__device__ __forceinline__ void async_copy_b128(const void* gsrc, void* ldst) {
  __builtin_amdgcn_global_load_async_to_lds_b128(
      (as1v4i*)(uintptr_t)gsrc,
      (as3v4i*)(uint32_t)(uintptr_t)ldst,
      0, 0);
}

__device__ __forceinline__ void async_wait_all() {
  __builtin_amdgcn_s_wait_asynccnt(0);
}

// ---------------------------------------------------------------------------
// Kernel 1: exact f32 WMMA projection.
//   Sp[b,n,do] = sum_k emb[b,n,k] * W1[do,k]       + b1[do]   (sender proj)
//   Rp[b,n,do] = sum_k emb[b,n,k] * W1[do,128+k]              (receiver proj)
// One wave computes one 16(n) x 16(do) tile of BOTH S and R (shared A operand).
// ---------------------------------------------------------------------------
__global__ __launch_bounds__(256) void proj_wmma_kernel(
    const float* __restrict__ emb, const float* __restrict__ W1,
    const float* __restrict__ b1, float* __restrict__ Sp, float* __restrict__ Rp)
{
  const int wave = threadIdx.x >> 5;
  const int lane = threadIdx.x & 31;
  const int tile = blockIdx.x * 8 + wave;      // 0..1471, grid sized exactly
  const int b  = tile / (NTILE * DTILE);
  const int r  = tile % (NTILE * DTILE);
  const int nt = r % NTILE;
  const int dt = r / NTILE;
  const int n0 = nt * 16;
  const int d0 = dt * 16;
  const int ln = lane & 15;   // A: row index M ; B: col index N (dout)
  const int kh = lane >> 4;   // which K-half of the 4-wide K block

  int arow = n0 + ln;
  if (arow >= NN) arow = NN - 1;               // clamp (stores are guarded)
  const float* aptr  = emb + ((size_t)b * NN + arow) * DD + 2 * kh;
  const float* bsptr = W1 + (size_t)(d0 + ln) * (2 * DD) + 2 * kh;  // W1[do, k]
  const float* brptr = bsptr + DD;                                  // W1[do, 128+k]

  v8f cs = {};
  v8f cr = {};
#pragma unroll 4
  for (int k = 0; k < DD; k += 4) {
    v2f a  = *(const v2f*)(aptr + k);
    v2f bs = *(const v2f*)(bsptr + k);
    v2f br = *(const v2f*)(brptr + k);
    cs = __builtin_amdgcn_wmma_f32_16x16x4_f32(false, a, false, bs,
                                               (short)0, cs, false, false);
    cr = __builtin_amdgcn_wmma_f32_16x16x4_f32(false, a, false, br,
                                               (short)0, cr, false, false);
  }

  const float bias = b1[d0 + ln];
#pragma unroll
  for (int v = 0; v < 8; ++v) {
    const int n = n0 + 8 * kh + v;
    if (n < NN) {
      const size_t off = ((size_t)b * NN + n) * DD + (size_t)(d0 + ln);
      Sp[off] = cs[v] + bias;
      Rp[off] = cr[v];
    }
  }
}

// ---------------------------------------------------------------------------
// Kernel 2: edge scores. ReLU inside the d-reduction makes this non-GEMM:
//   out[b,i,j] = relu( sum_d relu(Sp[b,j,d] + Rp[b,i,d]) * W2[d] + b2 )
// 32x32 (i,j) tile per block; S/R rows staged into LDS with async
// global->LDS b128 DMA (no VGPR round-trip), row stride 132 floats so row
// bases hit distinct banks; each thread owns a 2x2 output micro-tile
// (j, j+16) x (i, i+16) -> 4x operand reuse per ds_load_b128.
// ---------------------------------------------------------------------------
__global__ __launch_bounds__(256) void edge_score_kernel(
    const float* __restrict__ Sp, const float* __restrict__ Rp,
    const float* __restrict__ W2, const float* __restrict__ b2p,
    float* __restrict__ out)
{
  __shared__ float sS[TJ * LSTR];
  __shared__ float sR[TI * LSTR];
  __shared__ float sW[DD];

  const int b   = blockIdx.z;
  const int i0  = blockIdx.y * TI;
  const int j0  = blockIdx.x * TJ;
  const int tid = threadIdx.x;

  // W2: 128 floats = 32 b128 chunks, issued by lanes 0..31 of wave 0
  if (tid < 32) async_copy_b128(W2 + tid * 4, &sW[tid * 4]);

  // 64 tile rows x 128 floats = 2048 b128 chunks, 8 per thread.
  // Out-of-range rows are clamped (duplicate data; stores are guarded),
  // keeping EXEC uniform so every lane issues its async load.
#pragma unroll
  for (int idx = tid; idx < 64 * 32; idx += 256) {
    const int row = idx >> 5;
    const int c   = (idx & 31) * 4;
    if (row < TJ) {
      int j = j0 + row;
      if (j >= NN) j = NN - 1;
      async_copy_b128(Sp + ((size_t)b * NN + j) * DD + c, &sS[row * LSTR + c]);
    } else {
      int i = i0 + (row - TJ);
      if (i >= NN) i = NN - 1;
      async_copy_b128(Rp + ((size_t)b * NN + i) * DD + c,
                      &sR[(row - TJ) * LSTR + c]);
    }
  }
  async_wait_all();
  __syncthreads();

  const int jt = tid & 15;
  const int it = tid >> 4;
  const float* s0p = &sS[jt * LSTR];
  const float* s1p = &sS[(jt + 16) * LSTR];
  const float* r0p = &sR[it * LSTR];
  const float* r1p = &sR[(it + 16) * LSTR];

  float acc00 = 0.f, acc01 = 0.f, acc10 = 0.f, acc11 = 0.f;
#pragma unroll 4
  for (int d = 0; d < DD; d += 4) {
    const float4 w  = *(const float4*)(&sW[d]);
    const float4 s0 = *(const float4*)(s0p + d);
    const float4 s1 = *(const float4*)(s1p + d);
    const float4 r0 = *(const float4*)(r0p + d);
    const float4 r1 = *(const float4*)(r1p + d);
    acc00 += fmaxf(s0.x + r0.x, 0.f) * w.x + fmaxf(s0.y + r0.y, 0.f) * w.y
           + fmaxf(s0.z + r0.z, 0.f) * w.z + fmaxf(s0.w + r0.w, 0.f) * w.w;
    acc01 += fmaxf(s1.x + r0.x, 0.f) * w.x + fmaxf(s1.y + r0.y, 0.f) * w.y
           + fmaxf(s1.z + r0.z, 0.f) * w.z + fmaxf(s1.w + r0.w, 0.f) * w.w;
    acc10 += fmaxf(s0.x + r1.x, 0.f) * w.x + fmaxf(s0.y + r1.y, 0.f) * w.y
           + fmaxf(s0.z + r1.z, 0.f) * w.z + fmaxf(s0.w + r1.w, 0.f) * w.w;
    acc11 += fmaxf(s1.x + r1.x, 0.f) * w.x + fmaxf(s1.y + r1.y, 0.f) * w.y
           + fmaxf(s1.z + r1.z, 0.f) * w.z + fmaxf(s1.w + r1.w, 0.f) * w.w;
  }

  const float bias2 = b2p[0];
  const int ig0 = i0 + it, ig1 = ig0 + 16;
  const int jg0 = j0 + jt, jg1 = jg0 + 16;
  if (ig0 < NN) {
    const size_t rb = ((size_t)b * NN + ig0) * NN;
    if (jg0 < NN) out[rb + jg0] = fmaxf(acc00 + bias2, 0.f);
    if (jg1 < NN) out[rb + jg1] = fmaxf(acc01 + bias2, 0.f);
  }
  if (ig1 < NN) {
    const size_t rb = ((size_t)b * NN + ig1) * NN;
    if (jg0 < NN) out[rb + jg0] = fmaxf(acc10 + bias2, 0.f);
    if (jg1 < NN) out[rb + jg1] = fmaxf(acc11 + bias2, 0.f);
  }
}

// ---------------------------------------------------------------------------
extern "C" void kernel_launch(void* const* d_in, const int* in_sizes, int n_in,
                              void* d_out, int out_size, void* d_ws, size_t ws_size,
                              hipStream_t stream) {
  const float* emb = (const float*)d_in[0];  // [8,360,128]
  const float* W1  = (const float*)d_in[1];  // [128,256]
  const float* b1  = (const float*)d_in[2];  // [128]
  const float* W2  = (const float*)d_in[3];  // [1,128]
  const float* b2  = (const float*)d_in[4];  // [1]
  float* out = (float*)d_out;                // [8,360,360,1]

  float* Sp = (float*)d_ws;                       // [8,360,128]
  float* Rp = Sp + (size_t)BB * NN * DD;          // [8,360,128]

  // 8 batches * 23 n-tiles * 8 d-tiles = 1472 wave-tiles = 184 blocks x 8 waves
  proj_wmma_kernel<<<dim3(184), dim3(256), 0, stream>>>(emb, W1, b1, Sp, Rp);

  dim3 g2((NN + TJ - 1) / TJ, (NN + TI - 1) / TI, BB);  // 12 x 12 x 8
  edge_score_kernel<<<g2, dim3(256), 0, stream>>>(Sp, Rp, W2, b2, out);
}